// VitsConvFlow_30073361007318
// MI455X (gfx1250) — compile-verified
//
#include <hip/hip_runtime.h>
#include <math.h>

#define HIDDEN   192
#define T_LEN    2048
#define B_SZ     64
#define NBINS    10
#define KSZ      3
#define NLAYERS  3
#define TAILB    5.0f
#define MIN_BINF 0.001f
#define MIN_DERF 0.001f
#define LN_EPS   1e-5f
#define PROJ_OUT 29          // 3*NBINS - 1
#define INV_SQRT 0.07216878364870323f   // 1/sqrt(192)

#define DW_TILE  32          // t-tile width for depthwise kernel
#define DW_HALO  9           // max dilation (3^2)
#define DW_LDSW  (DW_TILE + 2 * DW_HALO)   // 50
#define DW_PITCH 52          // padded row pitch (floats)
#define DW_CPG   24          // channels per thread (192 / 8 groups)

typedef __attribute__((ext_vector_type(2))) float v2f;
typedef __attribute__((ext_vector_type(8))) float v8f;

__device__ __forceinline__ float gelu_exact(float x) {
    return 0.5f * x * (1.0f + erff(x * 0.7071067811865475f));
}

// ---------------------------------------------------------------------------
// Kernel 0: h[b,o,t] = conv_pre_w[o] * first[b,t] + conv_pre_b[o]
// ---------------------------------------------------------------------------
__global__ void pre_kernel(const float* __restrict__ inp,
                           const float* __restrict__ w,
                           const float* __restrict__ bias,
                           float* __restrict__ h, int total) {
    int idx = blockIdx.x * blockDim.x + threadIdx.x;
    if (idx >= total) return;
    int t = idx % T_LEN;
    int o = (idx / T_LEN) % HIDDEN;
    int b = idx / (T_LEN * HIDDEN);
    float first = inp[(size_t)(b * 2) * T_LEN + t];
    h[idx] = w[o] * first + bias[o];
}

// ---------------------------------------------------------------------------
// Kernel A: fused depthwise dilated conv + LayerNorm(channels) + GELU.
// Block = (b, 32-wide t tile); 256 threads = 8 channel-groups x 32 t.
// Input slab (192 ch x 50 t, mask applied, zero halo) staged in LDS with
// t-coalesced loads; 24 channels/thread kept in registers; per-t LN stats
// reduced across the 8 groups via LDS.
// ---------------------------------------------------------------------------
__global__ __launch_bounds__(256)
void dw_ln_gelu_kernel(const float* __restrict__ h,
                       const float* __restrict__ mask,
                       const float* __restrict__ dw_w,
                       const float* __restrict__ dw_b,
                       const float* __restrict__ g1,
                       const float* __restrict__ b1,
                       float* __restrict__ y1, int layer, int dil) {
    __shared__ float tile[HIDDEN * DW_PITCH];          // ~39.9 KB
    __shared__ float sums[8][DW_TILE];
    __shared__ float ssqs[8][DW_TILE];
    __shared__ float s_mu[DW_TILE], s_rs[DW_TILE];

    const int b  = blockIdx.y;
    const int t0 = blockIdx.x * DW_TILE;
    const int tid = threadIdx.x;

    // ---- stage input slab: h[b, c, t0-9 .. t0+40] * mask, zero-padded ----
    for (int idx = tid; idx < HIDDEN * DW_LDSW; idx += 256) {
        int c = idx / DW_LDSW;
        int j = idx - c * DW_LDSW;
        int tt = t0 - DW_HALO + j;
        float v = 0.0f;
        if (tt >= 0 && tt < T_LEN)
            v = h[((size_t)(b * HIDDEN + c)) * T_LEN + tt] * mask[b * T_LEN + tt];
        tile[c * DW_PITCH + j] = v;
    }
    __syncthreads();

    // ---- compute: thread = (csub, tl); 24 channels each ----
    const int tl   = tid & (DW_TILE - 1);
    const int csub = tid >> 5;          // 0..7
    float vreg[DW_CPG];
    float s = 0.0f, ss = 0.0f;
    #pragma unroll
    for (int k = 0; k < DW_CPG; ++k) {
        int c = csub * DW_CPG + k;
        const float* row = tile + c * DW_PITCH + tl + DW_HALO;
        float acc = dw_b[layer * HIDDEN + c];
        acc += dw_w[(layer * HIDDEN + c) * KSZ + 0] * row[-dil];
        acc += dw_w[(layer * HIDDEN + c) * KSZ + 1] * row[0];
        acc += dw_w[(layer * HIDDEN + c) * KSZ + 2] * row[dil];
        vreg[k] = acc;
        s  += acc;
        ss += acc * acc;
    }
    sums[csub][tl] = s;
    ssqs[csub][tl] = ss;
    __syncthreads();

    if (tid < DW_TILE) {
        float ts = 0.0f, tss = 0.0f;
        #pragma unroll
        for (int i = 0; i < 8; ++i) { ts += sums[i][tid]; tss += ssqs[i][tid]; }
        float mu = ts / (float)HIDDEN;
        s_mu[tid] = mu;
        s_rs[tid] = rsqrtf(tss / (float)HIDDEN - mu * mu + LN_EPS);
    }
    __syncthreads();

    const float mu = s_mu[tl];
    const float rs = s_rs[tl];
    #pragma unroll
    for (int k = 0; k < DW_CPG; ++k) {
        int c = csub * DW_CPG + k;
        float yv = (vreg[k] - mu) * rs * g1[layer * HIDDEN + c]
                   + b1[layer * HIDDEN + c];
        y1[((size_t)(b * HIDDEN + c)) * T_LEN + t0 + tl] = gelu_exact(yv);
    }
}

// ---------------------------------------------------------------------------
// Kernel B: pointwise 192x192 GEMM via fp32 WMMA + bias + LN(channels)
//           + GELU + residual add into h.
// block = 12 waves (384 thr); wave w owns M-tile [16w,16w+16) x N-tile 16.
// ---------------------------------------------------------------------------
__global__ __launch_bounds__(384)
void pw_wmma_kernel(const float* __restrict__ y1,
                    const float* __restrict__ pw_w,
                    const float* __restrict__ pw_b,
                    const float* __restrict__ g2,
                    const float* __restrict__ b2,
                    float* __restrict__ h, int layer) {
    __shared__ float tile[HIDDEN * 16];
    __shared__ float s_mu[16], s_rs[16];
    int b    = blockIdx.y;
    int t0   = blockIdx.x * 16;
    int wave = threadIdx.x >> 5;
    int lane = threadIdx.x & 31;
    int m0   = wave * 16;
    int row  = m0 + (lane & 15);
    int col  = t0 + (lane & 15);
    int kh   = (lane >> 4) * 2;      // lanes 0-15: K={k0,k0+1}; lanes 16-31: K={k0+2,k0+3}
    const float* W = pw_w + (size_t)layer * HIDDEN * HIDDEN;
    const float* X = y1 + (size_t)b * HIDDEN * T_LEN;

    v8f acc = {};
    for (int k0 = 0; k0 < HIDDEN; k0 += 4) {
        int kk = k0 + kh;
        v2f a, bb;
        a.x  = W[row * HIDDEN + kk];
        a.y  = W[row * HIDDEN + kk + 1];
        bb.x = X[(size_t)kk * T_LEN + col];
        bb.y = X[(size_t)(kk + 1) * T_LEN + col];
        acc = __builtin_amdgcn_wmma_f32_16x16x4_f32(false, a, false, bb,
                                                    (short)0, acc, false, false);
    }
    int mbase = m0 + (lane >> 4) * 8;
    int n = lane & 15;
    #pragma unroll
    for (int r = 0; r < 8; ++r) {
        int m = mbase + r;
        tile[m * 16 + n] = acc[r] + pw_b[layer * HIDDEN + m];
    }
    __syncthreads();
    if (threadIdx.x < 16) {
        float sum = 0.0f, ss = 0.0f;
        for (int m = 0; m < HIDDEN; ++m) {
            float v = tile[m * 16 + threadIdx.x];
            sum += v; ss += v * v;
        }
        float mu = sum / (float)HIDDEN;
        s_mu[threadIdx.x] = mu;
        s_rs[threadIdx.x] = rsqrtf(ss / (float)HIDDEN - mu * mu + LN_EPS);
    }
    __syncthreads();
    for (int idx = threadIdx.x; idx < HIDDEN * 16; idx += 384) {
        int m = idx >> 4, j = idx & 15;
        float v = (tile[idx] - s_mu[j]) * s_rs[j] * g2[layer * HIDDEN + m]
                  + b2[layer * HIDDEN + m];
        h[((size_t)(b * HIDDEN + m)) * T_LEN + t0 + j] += gelu_exact(v);
    }
}

// ---------------------------------------------------------------------------
// Kernel C: projection (29x192) GEMM via fp32 WMMA (M padded to 32).
// p[b,m,t] = (proj_w[m,:] . (h[b,:,t]*mask) + proj_b[m]) * mask
// block = 2 waves (64 thr) per (b, 16-column tile).
// ---------------------------------------------------------------------------
__global__ __launch_bounds__(64)
void proj_wmma_kernel(const float* __restrict__ h,
                      const float* __restrict__ mask,
                      const float* __restrict__ pj_w,
                      const float* __restrict__ pj_b,
                      float* __restrict__ p) {
    int b    = blockIdx.y;
    int t0   = blockIdx.x * 16;
    int wave = threadIdx.x >> 5;
    int lane = threadIdx.x & 31;
    int m0   = wave * 16;
    int row  = m0 + (lane & 15);
    int col  = t0 + (lane & 15);
    int kh   = (lane >> 4) * 2;
    float mk = mask[b * T_LEN + col];
    bool rok = (row < PROJ_OUT);

    v8f acc = {};
    for (int k0 = 0; k0 < HIDDEN; k0 += 4) {
        int kk = k0 + kh;
        v2f a, bb;
        a.x  = rok ? pj_w[row * HIDDEN + kk]     : 0.0f;
        a.y  = rok ? pj_w[row * HIDDEN + kk + 1] : 0.0f;
        bb.x = h[((size_t)(b * HIDDEN + kk)) * T_LEN + col] * mk;
        bb.y = h[((size_t)(b * HIDDEN + kk + 1)) * T_LEN + col] * mk;
        acc = __builtin_amdgcn_wmma_f32_16x16x4_f32(false, a, false, bb,
                                                    (short)0, acc, false, false);
    }
    int mbase = m0 + (lane >> 4) * 8;
    int n = lane & 15;
    #pragma unroll
    for (int r = 0; r < 8; ++r) {
        int m = mbase + r;
        if (m < PROJ_OUT)
            p[((size_t)(b * PROJ_OUT + m)) * T_LEN + t0 + n] = (acc[r] + pj_b[m]) * mk;
    }
}

// ---------------------------------------------------------------------------
// Kernel D: rational-quadratic spline + outputs + log_det reduction.
// thread per (b,t).
// ---------------------------------------------------------------------------
__device__ __forceinline__ void build_knots(const float* __restrict__ u,
                                            float* __restrict__ cw,
                                            float* __restrict__ wd) {
    float mx = u[0];
    #pragma unroll
    for (int k = 1; k < NBINS; ++k) mx = fmaxf(mx, u[k]);
    float e[NBINS];
    float s = 0.0f;
    #pragma unroll
    for (int k = 0; k < NBINS; ++k) { e[k] = expf(u[k] - mx); s += e[k]; }
    float inv = 1.0f / s;
    float run = 0.0f;
    cw[0] = -TAILB;
    #pragma unroll
    for (int k = 0; k < NBINS; ++k) {
        float w = MIN_BINF + (1.0f - MIN_BINF * (float)NBINS) * e[k] * inv;
        run += w;
        cw[k + 1] = 2.0f * TAILB * run - TAILB;
    }
    cw[NBINS] = TAILB;
    #pragma unroll
    for (int k = 0; k < NBINS; ++k) wd[k] = cw[k + 1] - cw[k];
}

__global__ __launch_bounds__(256)
void spline_kernel(const float* __restrict__ inp,
                   const float* __restrict__ mask,
                   const float* __restrict__ p,
                   float* __restrict__ out,
                   float* __restrict__ logdet) {
    __shared__ float red[256];
    int b = blockIdx.y;
    int t = blockIdx.x * 256 + threadIdx.x;

    float mk    = mask[b * T_LEN + t];
    float first = inp[(size_t)(b * 2) * T_LEN + t];
    float x     = inp[(size_t)(b * 2 + 1) * T_LEN + t];

    float uw[NBINS], uh[NBINS];
    #pragma unroll
    for (int k = 0; k < NBINS; ++k)
        uw[k] = p[((size_t)(b * PROJ_OUT + k)) * T_LEN + t] * INV_SQRT;
    #pragma unroll
    for (int k = 0; k < NBINS; ++k)
        uh[k] = p[((size_t)(b * PROJ_OUT + NBINS + k)) * T_LEN + t] * INV_SQRT;

    float der[NBINS + 1];
    der[0] = 1.0f;              // MIN_DER + softplus(log(exp(1-MIN_DER)-1)) == 1
    der[NBINS] = 1.0f;
    #pragma unroll
    for (int k = 1; k < NBINS; ++k) {
        float u  = p[((size_t)(b * PROJ_OUT + 2 * NBINS + k - 1)) * T_LEN + t];
        float sp = (u > 20.0f) ? u : log1pf(expf(u));
        der[k] = MIN_DERF + sp;
    }

    float cumw[NBINS + 1], widths[NBINS];
    float cumh[NBINS + 1], heights[NBINS];
    build_knots(uw, cumw, widths);
    build_knots(uh, cumh, heights);

    float xi = fminf(fmaxf(x, -TAILB), TAILB);
    float locLast = cumw[NBINS] + 1e-6f;
    int idx = -1;
    #pragma unroll
    for (int k = 0; k <= NBINS; ++k) {
        float loc = (k == NBINS) ? locLast : cumw[k];
        idx += (xi >= loc) ? 1 : 0;
    }
    idx = min(max(idx, 0), NBINS - 1);

    float in_cw = cumw[0], in_w = widths[0];
    float in_ch = cumh[0], in_h = heights[0];
    float in_der = der[0], in_der1 = der[1];
    #pragma unroll
    for (int k = 1; k < NBINS; ++k) {
        if (k == idx) {
            in_cw = cumw[k];  in_w = widths[k];
            in_ch = cumh[k];  in_h = heights[k];
            in_der = der[k];  in_der1 = der[k + 1];
        }
    }
    float in_d  = in_h / in_w;
    float theta = (xi - in_cw) / in_w;
    float t1m   = theta * (1.0f - theta);
    float i1    = in_der + in_der1 - 2.0f * in_d;
    float den   = in_d + i1 * t1m;
    float outv  = in_ch + in_h * (in_d * theta * theta + in_der * t1m) / den;
    float omt   = 1.0f - theta;
    float dnum  = in_d * in_d * (in_der1 * theta * theta + 2.0f * in_d * t1m
                                 + in_der * omt * omt);
    float lad   = logf(dnum) - 2.0f * logf(den);

    bool inside = (x >= -TAILB) && (x <= TAILB);
    float sec  = inside ? outv : x;
    float ladv = inside ? lad : 0.0f;

    out[(size_t)(b * 2)     * T_LEN + t] = first * mk;
    out[(size_t)(b * 2 + 1) * T_LEN + t] = sec * mk;

    red[threadIdx.x] = ladv * mk;
    __syncthreads();
    for (int s = 128; s > 0; s >>= 1) {
        if (threadIdx.x < s) red[threadIdx.x] += red[threadIdx.x + s];
        __syncthreads();
    }
    if (threadIdx.x == 0) atomicAdd(&logdet[b], red[0]);
}

__global__ void zero_logdet_kernel(float* __restrict__ ld) {
    if (threadIdx.x < B_SZ) ld[threadIdx.x] = 0.0f;
}

// ---------------------------------------------------------------------------
extern "C" void kernel_launch(void* const* d_in, const int* in_sizes, int n_in,
                              void* d_out, int out_size, void* d_ws, size_t ws_size,
                              hipStream_t stream) {
    const float* inputs     = (const float*)d_in[0];
    const float* mask       = (const float*)d_in[1];
    const float* conv_pre_w = (const float*)d_in[2];
    const float* conv_pre_b = (const float*)d_in[3];
    const float* dw_w       = (const float*)d_in[4];
    const float* dw_b       = (const float*)d_in[5];
    const float* pw_w       = (const float*)d_in[6];
    const float* pw_b       = (const float*)d_in[7];
    const float* ln1_g      = (const float*)d_in[8];
    const float* ln1_b      = (const float*)d_in[9];
    const float* ln2_g      = (const float*)d_in[10];
    const float* ln2_b      = (const float*)d_in[11];
    const float* pj_w       = (const float*)d_in[12];
    const float* pj_b       = (const float*)d_in[13];

    float* out = (float*)d_out;
    float* h   = (float*)d_ws;
    float* y1  = h + (size_t)B_SZ * HIDDEN * T_LEN;
    float* p   = y1;   // y1 is dead after last pointwise layer; reuse for proj output

    int total = B_SZ * HIDDEN * T_LEN;
    pre_kernel<<<(total + 255) / 256, 256, 0, stream>>>(inputs, conv_pre_w,
                                                        conv_pre_b, h, total);
    int dil = 1;
    for (int i = 0; i < NLAYERS; ++i) {
        dw_ln_gelu_kernel<<<dim3(T_LEN / DW_TILE, B_SZ), 256, 0, stream>>>(
            h, mask, dw_w, dw_b, ln1_g, ln1_b, y1, i, dil);
        pw_wmma_kernel<<<dim3(T_LEN / 16, B_SZ), 384, 0, stream>>>(
            y1, pw_w, pw_b, ln2_g, ln2_b, h, i);
        dil *= KSZ;
    }
    proj_wmma_kernel<<<dim3(T_LEN / 16, B_SZ), 64, 0, stream>>>(h, mask, pj_w,
                                                                pj_b, p);
    zero_logdet_kernel<<<1, 64, 0, stream>>>(out + (size_t)B_SZ * 2 * T_LEN);
    spline_kernel<<<dim3(T_LEN / 256, B_SZ), 256, 0, stream>>>(
        inputs, mask, p, out, out + (size_t)B_SZ * 2 * T_LEN);
}